// FlexibleWaveConv2d_24747601559877
// MI455X (gfx1250) — compile-verified
//
#include <hip/hip_runtime.h>

typedef float v2f __attribute__((ext_vector_type(2)));
typedef float v8f __attribute__((ext_vector_type(8)));

// Sizes (floats)
#define BAND_STRIDE 4194304ull          // 1024 pos * 32 b * 128 ch
#define A_TOTAL     16777216ull         // 4 bands
#define WT_TAP      16384ull            // 128*128
#define WT_BAND     4194304ull          // 256 taps * 16384

// ---------------------------------------------------------------------------
// Kernel 1: fused 2-level Haar DWT. Each level-2 band value is a +-0.25
// combination of a 4x4 block of x. Output layout A[band][pos][b][i] (i fast).
// grid (py=32, b=32), 128 threads (thread = channel c).
// ---------------------------------------------------------------------------
__global__ __launch_bounds__(128) void dwt2_kernel(const float* __restrict__ x,
                                                   float* __restrict__ A) {
  const int py = blockIdx.x;
  const int b  = blockIdx.y;
  const int c  = threadIdx.x;
  const float* xb = x + (((size_t)b * 128 + c) << 14);   // * 128*128
  for (int px = 0; px < 32; ++px) {
    float4 v0 = *(const float4*)&xb[(4 * py + 0) * 128 + 4 * px];
    float4 v1 = *(const float4*)&xb[(4 * py + 1) * 128 + 4 * px];
    float4 v2 = *(const float4*)&xb[(4 * py + 2) * 128 + 4 * px];
    float4 v3 = *(const float4*)&xb[(4 * py + 3) * 128 + 4 * px];
    float S00 = v0.x + v0.y + v1.x + v1.y;
    float S01 = v0.z + v0.w + v1.z + v1.w;
    float S10 = v2.x + v2.y + v3.x + v3.y;
    float S11 = v2.z + v2.w + v3.z + v3.w;
    float ll = 0.25f * (S00 + S01 + S10 + S11);
    float lh = 0.25f * (S00 - S01 + S10 - S11);
    float hl = 0.25f * (S00 + S01 - S10 - S11);
    float hh = 0.25f * (S00 - S01 - S10 + S11);
    size_t pos  = (size_t)py * 32 + px;
    size_t base = (pos * 32 + b) * 128 + c;
    A[base]                   = ll;
    A[base + BAND_STRIDE]     = lh;
    A[base + 2 * BAND_STRIDE] = hl;
    A[base + 3 * BAND_STRIDE] = hh;
  }
}

// ---------------------------------------------------------------------------
// Kernel 2: weight transpose (i,o,ty,tx) -> Wt[band][tap][i][o] so every
// bilinear tap is a contiguous 64KB matrix. grid (i=128, band=4), 256 thr.
// ---------------------------------------------------------------------------
__global__ __launch_bounds__(256) void wtrans_kernel(const float* __restrict__ w1,
                                                     const float* __restrict__ w2,
                                                     const float* __restrict__ w3,
                                                     const float* __restrict__ w4,
                                                     float* __restrict__ Wt) {
  __shared__ float sT[32 * 257];                         // pad 257 -> conflict-free
  const int band = blockIdx.y;
  const int i    = blockIdx.x;
  const int tid  = threadIdx.x;
  const float* w = (band == 0) ? w1 : (band == 1) ? w2 : (band == 2) ? w3 : w4;

  for (int ob = 0; ob < 4; ++ob) {                       // 32 o-columns at a time
    if (ob) __syncthreads();
    const float4* src = (const float4*)(w + (size_t)i * 32768 + (size_t)ob * 8192);
    for (int it = 0; it < 8; ++it) {
      int l4 = tid + 256 * it;                           // 0..2047 float4s
      float4 v = src[l4];
      int o  = l4 >> 6;                                  // (4*l4)/256
      int t0 = (l4 & 63) * 4;
      float* d = &sT[o * 257 + t0];
      d[0] = v.x; d[1] = v.y; d[2] = v.z; d[3] = v.w;
    }
    __syncthreads();
    int o = tid & 31, w8 = tid >> 5;
    for (int j = 0; j < 32; ++j) {
      int t = w8 + 8 * j;
      Wt[(((size_t)band * 256 + t) * 128 + i) * 128 + ob * 32 + o] = sT[o * 257 + t];
    }
  }
}

// ---------------------------------------------------------------------------
// Kernel 3: per (band, position) GEMM  out[32,128] = A[32,128] @ Wp[128,128]
// Wp built in LDS from 4 bilinear taps (L2-resident).  FP32 WMMA 16x16x4.
// grid (pos=1024, band=4), 256 threads = 8 waves, each wave -> 2 C tiles.
// Results overwrite the A block in place.
// ---------------------------------------------------------------------------
__global__ __launch_bounds__(256) void wave_gemm_kernel(const float* __restrict__ Wt,
                                                        float* __restrict__ AG) {
  __shared__ float sA[32 * 132];                         // pad 132: conflict-free A frags
  __shared__ float sBT[128 * 68];                        // B transposed [o][k], pad 68

  const int pos  = blockIdx.x;
  const int band = blockIdx.y;
  const int py = pos >> 5, px = pos & 31;
  const int tid = threadIdx.x;

  // Bilinear taps (half-pixel centers, scale 2, edge clamp)
  int ky = py >> 1, iy0, iy1; float cy0;
  if ((py & 1) == 0) { iy0 = ky - 1; iy1 = ky;     cy0 = 0.25f; }
  else               { iy0 = ky;     iy1 = ky + 1; cy0 = 0.75f; }
  float cy1 = 1.0f - cy0;
  iy0 = iy0 < 0 ? 0 : iy0;  iy1 = iy1 > 15 ? 15 : iy1;
  int kx = px >> 1, ix0, ix1; float cx0;
  if ((px & 1) == 0) { ix0 = kx - 1; ix1 = kx;     cx0 = 0.25f; }
  else               { ix0 = kx;     ix1 = kx + 1; cx0 = 0.75f; }
  float cx1 = 1.0f - cx0;
  ix0 = ix0 < 0 ? 0 : ix0;  ix1 = ix1 > 15 ? 15 : ix1;

  const float c00 = cy0 * cx0, c01 = cy0 * cx1, c10 = cy1 * cx0, c11 = cy1 * cx1;
  const float* wb  = Wt + (size_t)band * WT_BAND;
  const float* w00 = wb + (size_t)(iy0 * 16 + ix0) * WT_TAP;
  const float* w01 = wb + (size_t)(iy0 * 16 + ix1) * WT_TAP;
  const float* w10 = wb + (size_t)(iy1 * 16 + ix0) * WT_TAP;
  const float* w11 = wb + (size_t)(iy1 * 16 + ix1) * WT_TAP;

  float* ablk = AG + ((size_t)band * 1024 + pos) * 4096;

  // stage A (32x128) into LDS
  {
    const float4* src = (const float4*)ablk;
    int b0 = tid >> 5, i4 = tid & 31;
    for (int j = 0; j < 4; ++j) {
      int b = b0 + j * 8;
      float4 v = src[b * 32 + i4];
      *(float4*)&sA[b * 132 + i4 * 4] = v;
    }
  }

  const int wave = tid >> 5, lane = tid & 31;
  const int half = lane >> 4, l = lane & 15;
  const int mtile = wave & 1;
  const int n0 = wave >> 1, n1 = n0 + 4;

  v8f acc0 = {}, acc1 = {};

  const int oo = tid & 127;   // staging: this thread's o column
  const int hh = tid >> 7;    // staging: k half within phase

  for (int p = 0; p < 2; ++p) {                          // K split: [0,64) [64,128)
    if (p) __syncthreads();                              // phase-0 reads done
    for (int kk = 0; kk < 32; ++kk) {
      int krow = p * 64 + hh * 32 + kk;
      size_t off = (size_t)krow * 128 + oo;
      float v = c00 * w00[off] + c01 * w01[off] + c10 * w10[off] + c11 * w11[off];
      sBT[oo * 68 + hh * 32 + kk] = v;                   // B^T [o][k_local]
    }
    __syncthreads();
#pragma unroll
    for (int kk = 0; kk < 64; kk += 4) {
      // A frag: lanes 0-15 hold K=k0,k0+1 of row M=l; lanes 16-31 hold k0+2,k0+3
      v2f a  = *(const v2f*)&sA[(mtile * 16 + l) * 132 + p * 64 + kk + 2 * half];
      // B frag: VGPR0 = row K=k0+2*half, VGPR1 = row K=k0+2*half+1, col N=l
      v2f bv0 = *(const v2f*)&sBT[(n0 * 16 + l) * 68 + kk + 2 * half];
      v2f bv1 = *(const v2f*)&sBT[(n1 * 16 + l) * 68 + kk + 2 * half];
      acc0 = __builtin_amdgcn_wmma_f32_16x16x4_f32(false, a, false, bv0,
                                                   (short)0, acc0, false, false);
      acc1 = __builtin_amdgcn_wmma_f32_16x16x4_f32(false, a, false, bv1,
                                                   (short)0, acc1, false, false);
    }
  }

  // C layout: VGPR r -> M = r (lanes 0-15) / r+8 (lanes 16-31), N = l
  for (int r = 0; r < 8; ++r) {
    int brow = mtile * 16 + r + half * 8;
    ablk[brow * 128 + n0 * 16 + l] = acc0[r];
    ablk[brow * 128 + n1 * 16 + l] = acc1[r];
  }
}

// ---------------------------------------------------------------------------
// Kernel 4: level-2 idwt + zero-detail idwt (= x2 upsample * 0.5), fused.
// grid (py=32, b=32), 256 threads; float4 coalesced stores of the 268MB out.
// ---------------------------------------------------------------------------
__global__ __launch_bounds__(256) void idwt_up_kernel(const float* __restrict__ G,
                                                      float* __restrict__ out) {
  const int py = blockIdx.x, b = blockIdx.y;
  const int wave = threadIdx.x >> 5, px = threadIdx.x & 31;
  const size_t pos   = (size_t)py * 32 + px;
  const size_t gbase = (pos * 32 + b) * 128;
  for (int it = 0; it < 16; ++it) {
    int o = wave * 16 + it;
    float ll = G[gbase + o];
    float lh = G[gbase + BAND_STRIDE + o];
    float hl = G[gbase + 2 * BAND_STRIDE + o];
    float hh = G[gbase + 3 * BAND_STRIDE + o];
    float x00 = 0.5f * (ll + lh + hl + hh);
    float x01 = 0.5f * (ll - lh + hl - hh);
    float x10 = 0.5f * (ll + lh - hl - hh);
    float x11 = 0.5f * (ll - lh - hl + hh);
    float* orow = out + (((size_t)b * 128 + o) * 128 + 4 * py) * 128 + 4 * px;
    float4 r0 = {0.5f * x00, 0.5f * x00, 0.5f * x01, 0.5f * x01};
    float4 r1 = {0.5f * x10, 0.5f * x10, 0.5f * x11, 0.5f * x11};
    *(float4*)&orow[0]   = r0;
    *(float4*)&orow[128] = r0;
    *(float4*)&orow[256] = r1;
    *(float4*)&orow[384] = r1;
  }
}

extern "C" void kernel_launch(void* const* d_in, const int* in_sizes, int n_in,
                              void* d_out, int out_size, void* d_ws, size_t ws_size,
                              hipStream_t stream) {
  (void)in_sizes; (void)n_in; (void)out_size; (void)ws_size;
  const float* x  = (const float*)d_in[0];
  const float* w1 = (const float*)d_in[1];
  const float* w2 = (const float*)d_in[2];
  const float* w3 = (const float*)d_in[3];
  const float* w4 = (const float*)d_in[4];
  float* A  = (float*)d_ws;          // 67.1 MB: band coeffs, then GEMM results
  float* Wt = A + A_TOTAL;           // 67.1 MB: tap-major transposed weights
  float* out = (float*)d_out;

  dwt2_kernel   <<<dim3(32, 32),   128, 0, stream>>>(x, A);
  wtrans_kernel <<<dim3(128, 4),   256, 0, stream>>>(w1, w2, w3, w4, Wt);
  wave_gemm_kernel<<<dim3(1024, 4), 256, 0, stream>>>(Wt, A);
  idwt_up_kernel<<<dim3(32, 32),   256, 0, stream>>>(A, out);
}